// ScheduleMaxBPLayerV2_14516989461231
// MI455X (gfx1250) — compile-verified
//
#include <hip/hip_runtime.h>

#define LN_ZERO (-100000000000.0f)
#define NEG_BIG (-3.0e38f)

typedef __attribute__((ext_vector_type(2))) float v2f_t;
typedef __attribute__((ext_vector_type(8))) float v8f_t;

// Wave-internal LDS ordering fence: LDS ops from one wave complete in order
// (DScnt); this waits them out and stops compiler reordering around it.
__device__ __forceinline__ void wave_lds_fence() {
    asm volatile("s_wait_dscnt 0" ::: "memory");
}

__global__ void zero_ws_kernel(float* __restrict__ p, int n) {
    int i = blockIdx.x * blockDim.x + threadIdx.x;
    if (i < n) p[i] = 0.0f;
}

// vb[v,c] += msgs[e,c] for every edge e with edge_v[e]==v  (segment_sum)
__global__ void scatter_add_kernel(const float* __restrict__ msgs,
                                   const int* __restrict__ edge_v,
                                   float* __restrict__ vb, int EC) {
    int i = blockIdx.x * blockDim.x + threadIdx.x;
    if (i < EC) {
        int e = i >> 1, c = i & 1;
        unsafeAtomicAdd(&vb[(size_t)edge_v[e] * 2 + c], msgs[i]);
    }
}

__global__ void vb_out_kernel(const float* __restrict__ vb_raw,
                              const unsigned char* __restrict__ vb_mask,
                              float* __restrict__ vb_out, int VC) {
    int i = blockIdx.x * blockDim.x + threadIdx.x;
    if (i < VC) {
        float x = vb_mask[i] ? LN_ZERO : vb_raw[i];
        vb_out[i] = fmaxf(x, LN_ZERO);
    }
}

// ---- shared helper: build v2f tile in LDS, then fb tile via 2x WMMA f32 ----
// Tile: 16 factors per wave. v2fL layout: [16 factors][d*2+c] = 128 floats.
// fb = pot + A(16x8-as-2xK4) x B(8x16 0/1 selector), f32 WMMA 16x16x4.
__device__ __forceinline__ void build_v2f_tile(
    float* v2fL, int fbase, int F, int lane,
    const float* __restrict__ msgs, const int* __restrict__ edge_v,
    const float* __restrict__ vb_raw,
    const unsigned char* __restrict__ vb_mask,
    const unsigned char* __restrict__ v2f_mask) {
#pragma unroll
    for (int j = 0; j < 4; ++j) {
        int idx = lane * 4 + j;          // 0..127
        int fl  = idx >> 3;              // local factor 0..15
        int rem = idx & 7;               // d*2+c
        int d = rem >> 1, c = rem & 1;
        int fg = min(fbase + fl, F - 1);
        int e  = fg * 4 + d;
        int vi = edge_v[e] * 2 + c;
        float vbv = vb_mask[vi] ? LN_ZERO : vb_raw[vi];
        vbv = fmaxf(vbv, LN_ZERO);
        float t = vbv - msgs[(size_t)e * 2 + c];
        t = v2f_mask[(size_t)e * 2 + c] ? LN_ZERO : t;
        v2fL[idx] = fmaxf(t, LN_ZERO);
    }
    wave_lds_fence();
}

__device__ __forceinline__ v8f_t fb_tile_wmma(
    const float* v2fL, int fbase, int F, int lane,
    const float* __restrict__ pot,
    const int* __restrict__ pot_mask) {
    const int Mlane = lane & 15;   // A-matrix row
    const int kh    = lane >> 4;   // K-half select (0/1)
    const int n     = lane & 15;   // B/C/D column (state)

    v8f_t acc;
#pragma unroll
    for (int r = 0; r < 8; ++r) {  // C layout: VGPR r -> M = r + 8*kh
        int fg = min(fbase + r + 8 * kh, F - 1);
        acc[r] = pot[(size_t)fg * 16 + n];
    }
#pragma unroll
    for (int q = 0; q < 2; ++q) {  // two K=4 chunks of the (d,c) axis
        v2f_t a, b;
        // A 16x4 layout: VGPR r holds K = r + 2*kh for row Mlane
        a.x = v2fL[Mlane * 8 + q * 4 + 2 * kh + 0];
        a.y = v2fL[Mlane * 8 + q * 4 + 2 * kh + 1];
        // B 4x16 selector: B[k][n] = (digit(n,d)==c), j = q*4 + (r + 2*kh)
        {
            int j0 = q * 4 + 0 + 2 * kh, d0 = j0 >> 1, c0 = j0 & 1;
            int j1 = q * 4 + 1 + 2 * kh, d1 = j1 >> 1, c1 = j1 & 1;
            b.x = (((n >> (3 - d0)) & 1) == c0) ? 1.0f : 0.0f;
            b.y = (((n >> (3 - d1)) & 1) == c1) ? 1.0f : 0.0f;
        }
        acc = __builtin_amdgcn_wmma_f32_16x16x4_f32(
            false, a, false, b, (short)0, acc, false, false);
    }
    // factor-potential mask clamp
#pragma unroll
    for (int r = 0; r < 8; ++r) {
        int fg = min(fbase + r + 8 * kh, F - 1);
        float x = (pot_mask[(size_t)fg * 16 + n] == 1) ? LN_ZERO : acc[r];
        acc[r] = fmaxf(x, LN_ZERO);
    }
    return acc;
}

// Pass 1: vb1 -> v2f -> fb -> max-marginalize -> logsumexp -> f2v (output)
__global__ __launch_bounds__(256) void factor_pass1_kernel(
    const float* __restrict__ msgs, const float* __restrict__ pot,
    const int* __restrict__ edge_v, const float* __restrict__ vb1,
    const int* __restrict__ pot_mask,
    const unsigned char* __restrict__ vb_mask,
    const unsigned char* __restrict__ v2f_mask,
    const unsigned char* __restrict__ f2v_mask,
    float* __restrict__ out_f2v, int F) {
    __shared__ float sv2f[8][128];
    __shared__ float sfb[8][256];
    const int wlocal = threadIdx.x >> 5;
    const int lane   = threadIdx.x & 31;
    const int fbase  = (blockIdx.x * 8 + wlocal) * 16;
    if (fbase >= F) return;                 // uniform per wave
    float* v2fL = sv2f[wlocal];
    float* fbL  = sfb[wlocal];

    build_v2f_tile(v2fL, fbase, F, lane, msgs, edge_v, vb1, vb_mask, v2f_mask);
    v8f_t fb = fb_tile_wmma(v2fL, fbase, F, lane, pot, pot_mask);

    const int kh = lane >> 4, n = lane & 15;
#pragma unroll
    for (int r = 0; r < 8; ++r) fbL[(r + 8 * kh) * 16 + n] = fb[r];
    wave_lds_fence();

    // 64 edges per tile, 2 per lane: max over the 8 states with digit==c,
    // subtract v2f, per-row (C=2) stable logsumexp, mask clamp, store.
#pragma unroll
    for (int t = 0; t < 2; ++t) {
        int ew = lane + t * 32;
        int fl = ew >> 2;
        int d  = ew & 3;
        float m0 = NEG_BIG, m1 = NEG_BIG;
#pragma unroll
        for (int k = 0; k < 16; ++k) {
            float x = fbL[fl * 16 + k];
            if (((k >> (3 - d)) & 1) == 0) m0 = fmaxf(m0, x);
            else                           m1 = fmaxf(m1, x);
        }
        float a0 = m0 - v2fL[fl * 8 + d * 2 + 0];
        float a1 = m1 - v2fL[fl * 8 + d * 2 + 1];
        // global-max shift cancels in x - logsumexp(x); do stable row lse
        float mx  = fmaxf(a0, a1);
        float lse = mx + __logf(__expf(a0 - mx) + __expf(a1 - mx));
        int fg = fbase + fl;
        if (fg < F) {
            size_t e2 = (size_t)(fg * 4 + d) * 2;
            float o0 = a0 - lse, o1 = a1 - lse;
            o0 = f2v_mask[e2 + 0] ? LN_ZERO : o0;
            o1 = f2v_mask[e2 + 1] ? LN_ZERO : o1;
            out_f2v[e2 + 0] = fmaxf(o0, LN_ZERO);
            out_f2v[e2 + 1] = fmaxf(o1, LN_ZERO);
        }
    }
}

// Pass 2: vb2 + f2v -> v2f2 -> fb2 (output)
__global__ __launch_bounds__(256) void factor_pass2_kernel(
    const float* __restrict__ f2v, const float* __restrict__ pot,
    const int* __restrict__ edge_v, const float* __restrict__ vb2,
    const int* __restrict__ pot_mask,
    const unsigned char* __restrict__ vb_mask,
    const unsigned char* __restrict__ v2f_mask,
    float* __restrict__ out_fb, int F) {
    __shared__ float sv2f[8][128];
    const int wlocal = threadIdx.x >> 5;
    const int lane   = threadIdx.x & 31;
    const int fbase  = (blockIdx.x * 8 + wlocal) * 16;
    if (fbase >= F) return;
    float* v2fL = sv2f[wlocal];

    build_v2f_tile(v2fL, fbase, F, lane, f2v, edge_v, vb2, vb_mask, v2f_mask);
    v8f_t fb = fb_tile_wmma(v2fL, fbase, F, lane, pot, pot_mask);

    const int kh = lane >> 4, n = lane & 15;
#pragma unroll
    for (int r = 0; r < 8; ++r) {
        int fg = fbase + r + 8 * kh;
        if (fg < F) out_fb[(size_t)fg * 16 + n] = fb[r];
    }
}

extern "C" void kernel_launch(void* const* d_in, const int* in_sizes, int n_in,
                              void* d_out, int out_size, void* d_ws, size_t ws_size,
                              hipStream_t stream) {
    const float* msgs          = (const float*)d_in[0];          // [E,2]
    const float* pot           = (const float*)d_in[1];          // [F,16]
    const int*   edge_idx      = (const int*)d_in[2];            // [2,E]
    // d_in[3], d_in[4]: static scatter patterns -- derived analytically, unread
    const int*   pot_mask      = (const int*)d_in[5];            // [F,16] i32
    const unsigned char* f2v_mask = (const unsigned char*)d_in[6]; // [E,2] bool
    const unsigned char* v2f_mask = (const unsigned char*)d_in[7]; // [E,2] bool
    const unsigned char* vb_mask  = (const unsigned char*)d_in[8]; // [V,2] bool

    const int E = in_sizes[0] / 2;
    const int F = in_sizes[1] / 16;
    const int V = in_sizes[8] / 2;
    const int* edge_v = edge_idx + E;   // second row of facToVar_edge_idx

    float* vb1 = (float*)d_ws;                 // [V,2] raw sums, pass 1
    float* vb2 = vb1 + (size_t)V * 2;          // [V,2] raw sums, pass 2

    float* out_f2v = (float*)d_out;            // [E,2]
    float* out_vb  = out_f2v + (size_t)E * 2;  // [V,2]
    float* out_fb  = out_vb + (size_t)V * 2;   // [F,16]

    const int T = 256;
    const int nz = V * 4;                       // vb1+vb2 contiguous
    zero_ws_kernel<<<(nz + T - 1) / T, T, 0, stream>>>(vb1, nz);

    scatter_add_kernel<<<(E * 2 + T - 1) / T, T, 0, stream>>>(msgs, edge_v, vb1, E * 2);

    const int nwaves  = (F + 15) / 16;
    const int nblocks = (nwaves + 7) / 8;
    factor_pass1_kernel<<<nblocks, T, 0, stream>>>(
        msgs, pot, edge_v, vb1, pot_mask, vb_mask, v2f_mask, f2v_mask, out_f2v, F);

    scatter_add_kernel<<<(E * 2 + T - 1) / T, T, 0, stream>>>(out_f2v, edge_v, vb2, E * 2);

    vb_out_kernel<<<(V * 2 + T - 1) / T, T, 0, stream>>>(vb2, vb_mask, out_vb, V * 2);

    factor_pass2_kernel<<<nblocks, T, 0, stream>>>(
        out_f2v, pot, edge_v, vb2, pot_mask, vb_mask, v2f_mask, out_fb, F);
}